// RelGraphConvLowMem_7009386627524
// MI455X (gfx1250) — compile-verified
//
#include <hip/hip_runtime.h>

#define NUM_RELS  64
#define EPR       16384
#define E_TOTAL   (NUM_RELS * EPR)
#define IN_FEAT   128
#define OUT_FEAT  128
#define NUM_BASES 16
#define N_NODES   100000
#define M_TILES   (N_NODES / 16)   // 6250, exact

typedef float v2f __attribute__((ext_vector_type(2)));
typedef float v8f __attribute__((ext_vector_type(8)));

// sched_group_barrier masks
#define SCHED_DS_READ 0x100
#define SCHED_MATRIX  0x008

// ---------------------------------------------------------------------------
// LDS layout (floats):
//   sB   : 16384  floats (64KB)  -- B matrix in WMMA fragment order
//   sA   : 16896  floats (66KB)  -- 8 waves * 16 rows * 132 (padded) floats
//   sNorm: 128 floats, sDst: 128 ints (edge kernel only)
//
// sB fragment order: float2 at sB2[(ntile*32 + kstep)*32 + lane] holds
//   .x = B[kstep*4 + 2*(lane>=16) + 0][ntile*16 + (lane&15)]
//   .y = B[kstep*4 + 2*(lane>=16) + 1][ntile*16 + (lane&15)]
// which is exactly V_WMMA_F32_16X16X4_F32's B operand layout, so the compute
// loop reads it with conflict-free ds_load_b64 (banks 2*lane, 2*lane+1).
// ---------------------------------------------------------------------------

// Wave w stages N-tile w directly in fragment order. Each global load is two
// coalesced 64B segments (rows k and k+2 of a 16-column slice); each LDS
// store is a conflict-free b64 at word 2*lane.
__device__ __forceinline__ void stage_B_direct(const float* __restrict__ src,
                                               float* __restrict__ sB,
                                               int wave, int lane) {
  const int koff = (lane >> 4) << 1;                    // 0 or 2
  const float* colBase = src + (wave << 4) + (lane & 15);
  float2* dst = ((float2*)sB) + (wave << 10) + lane;    // (wave*32+ks)*32+lane
#pragma unroll
  for (int ks = 0; ks < 32; ++ks) {
    int k = (ks << 2) + koff;
    float2 v;
    v.x = colBase[(size_t)k * OUT_FEAT];
    v.y = colBase[(size_t)(k + 1) * OUT_FEAT];
    dst[ks << 5] = v;
  }
}

// ---------------------------------------------------------------------------
__global__ void zero_out_kernel(float4* __restrict__ p, int n4) {
  int i = blockIdx.x * blockDim.x + threadIdx.x;
  if (i < n4) p[i] = make_float4(0.f, 0.f, 0.f, 0.f);
}

// W[r,i,o] = sum_b w_comp[r,b] * weight[b,i,o]
__global__ void basis_kernel(const float* __restrict__ w_comp,
                             const float* __restrict__ weight,
                             float* __restrict__ W) {
  int idx = blockIdx.x * 256 + threadIdx.x;        // 0 .. 64*16384-1
  int r  = idx >> 14;
  int io = idx & 16383;
  float s = 0.f;
#pragma unroll
  for (int b = 0; b < NUM_BASES; ++b)
    s = fmaf(w_comp[r * NUM_BASES + b], weight[b * 16384 + io], s);
  W[idx] = s;
}

// ---------------------------------------------------------------------------
// Main fused edge kernel: gather -> grouped GEMM (f32 WMMA) -> norm -> atomic
// scatter. One block = 128 edges of one relation (8 waves x 16-edge tile).
// ---------------------------------------------------------------------------
__global__ __launch_bounds__(256) void rgcn_edge_kernel(
    const float* __restrict__ feat, const float* __restrict__ Wbuf,
    const float* __restrict__ norm, const long long* __restrict__ src,
    const long long* __restrict__ dst, float* __restrict__ out) {
  extern __shared__ float smem[];
  float* sB    = smem;                 // 16384 floats
  float* sA    = smem + 16384;         // 16896 floats
  float* sNorm = sA + 16896;           // 128 floats
  int*   sDst  = (int*)(sNorm + 128);  // 128 ints

  const int tid  = threadIdx.x;
  const int wave = tid >> 5;
  const int lane = tid & 31;
  const int rel       = blockIdx.x >> 7;   // 128 blocks per relation
  const int edge_base = blockIdx.x << 7;   // 128 edges per block

  // B = W[rel], staged in WMMA fragment order (wave w -> ntile w)
  stage_B_direct(Wbuf + (size_t)rel * (IN_FEAT * OUT_FEAT), sB, wave, lane);

  // per-edge metadata (lanes 0..15 of each wave)
  int myNode = 0;
  {
    long long e = (long long)edge_base + wave * 16 + (lane & 15);
    if (lane < 16) {
      myNode = (int)src[e];
      sNorm[wave * 16 + lane] = norm[e];
      sDst[wave * 16 + lane]  = (int)dst[e];
    }
  }
  // gather this wave's 16x128 A tile (one feat row per iteration, coalesced)
#pragma unroll
  for (int r = 0; r < 16; ++r) {
    int node = __shfl(myNode, r, 32);
    float4 v = *(const float4*)(feat + (size_t)node * IN_FEAT + (lane << 2));
    *(float4*)(sA + (wave * 16 + r) * 132 + (lane << 2)) = v;
  }
  __syncthreads();

  const int m    = lane & 15;
  const int koff = (lane >> 4) << 1;                  // A lane K offset (0 or 2)
  const float* aRow = sA + (wave * 16 + m) * 132 + koff;
  const float2* sB2 = (const float2*)sB;

  // Hoist all A fragments into registers (conflict-free float2 reads).
  float2 aFrag[32];
#pragma unroll
  for (int ks = 0; ks < 32; ++ks)
    aFrag[ks] = *(const float2*)(aRow + (ks << 2));

  // Hoist ntile-invariant epilogue metadata: per-row norm and dest pointer.
  float  normR[8];
  float* outP[8];
#pragma unroll
  for (int j = 0; j < 8; ++j) {
    int row  = j + ((lane >> 4) << 3);
    normR[j] = sNorm[wave * 16 + row];
    outP[j]  = out + (size_t)sDst[wave * 16 + row] * OUT_FEAT + m;
  }

#pragma unroll 1
  for (int ntile = 0; ntile < 8; ++ntile) {
    const float2* bp = sB2 + (ntile << 10) + lane;
    float2 bf[32];
#pragma unroll
    for (int ks = 0; ks < 32; ++ks) bf[ks] = bp[ks << 5];

    v8f acc = {0.f, 0.f, 0.f, 0.f, 0.f, 0.f, 0.f, 0.f};
#pragma unroll
    for (int ks = 0; ks < 32; ++ks) {
      v2f A = {aFrag[ks].x, aFrag[ks].y};
      v2f B = {bf[ks].x, bf[ks].y};
      acc = __builtin_amdgcn_wmma_f32_16x16x4_f32(false, A, false, B,
                                                  (short)0, acc, false, false);
    }
    // Force schedule: all DS reads issue first, then the WMMA burst, so only
    // the first load's LDS latency is exposed (descending dscnt waits).
    __builtin_amdgcn_sched_group_barrier(SCHED_DS_READ, 32, 0);
    __builtin_amdgcn_sched_group_barrier(SCHED_MATRIX, 32, 0);

    // D layout: vgpr j -> row (j + 8*(lane>=16)), col = ntile*16 + (lane&15)
#pragma unroll
    for (int j = 0; j < 8; ++j)
      atomicAdd(outP[j] + (ntile << 4), acc[j] * normR[j]);
  }
}

// ---------------------------------------------------------------------------
// Self-loop GEMM + bias + ReLU, fused with the accumulated scatter result
// (in-place on d_out). Node tiles are contiguous, no gather needed.
// ---------------------------------------------------------------------------
__global__ __launch_bounds__(256) void rgcn_loop_kernel(
    const float* __restrict__ feat, const float* __restrict__ loopW,
    const float* __restrict__ bias, float* __restrict__ out) {
  extern __shared__ float smem[];
  float* sB = smem;            // 16384 floats
  float* sA = smem + 16384;    // 16896 floats

  const int tid  = threadIdx.x;
  const int wave = tid >> 5;
  const int lane = tid & 31;
  const int tile = blockIdx.x * 8 + wave;   // 16-node tile per wave
  const bool valid = tile < M_TILES;        // wave-uniform

  stage_B_direct(loopW, sB, wave, lane);

  if (valid) {
    const float* fbase = feat + (size_t)tile * 16 * IN_FEAT;
#pragma unroll
    for (int r = 0; r < 16; ++r) {
      float4 v = ((const float4*)(fbase + r * IN_FEAT))[lane];
      *(float4*)(sA + (wave * 16 + r) * 132 + (lane << 2)) = v;
    }
  }
  __syncthreads();
  if (!valid) return;

  const int m    = lane & 15;
  const int koff = (lane >> 4) << 1;
  const float* aRow = sA + (wave * 16 + m) * 132 + koff;
  const float2* sB2 = (const float2*)sB;

  float2 aFrag[32];
#pragma unroll
  for (int ks = 0; ks < 32; ++ks)
    aFrag[ks] = *(const float2*)(aRow + (ks << 2));

  // ntile-invariant output row pointers
  float* outP[8];
#pragma unroll
  for (int j = 0; j < 8; ++j) {
    int row = j + ((lane >> 4) << 3);
    outP[j] = out + (size_t)(tile * 16 + row) * OUT_FEAT + m;
  }

#pragma unroll 1
  for (int ntile = 0; ntile < 8; ++ntile) {
    const float2* bp = sB2 + (ntile << 10) + lane;
    float2 bf[32];
#pragma unroll
    for (int ks = 0; ks < 32; ++ks) bf[ks] = bp[ks << 5];

    v8f acc = {0.f, 0.f, 0.f, 0.f, 0.f, 0.f, 0.f, 0.f};
#pragma unroll
    for (int ks = 0; ks < 32; ++ks) {
      v2f A = {aFrag[ks].x, aFrag[ks].y};
      v2f B = {bf[ks].x, bf[ks].y};
      acc = __builtin_amdgcn_wmma_f32_16x16x4_f32(false, A, false, B,
                                                  (short)0, acc, false, false);
    }
    __builtin_amdgcn_sched_group_barrier(SCHED_DS_READ, 32, 0);
    __builtin_amdgcn_sched_group_barrier(SCHED_MATRIX, 32, 0);

    float bv = bias[(ntile << 4) + m];
#pragma unroll
    for (int j = 0; j < 8; ++j) {
      float* p = outP[j] + (ntile << 4);
      *p = fmaxf(acc[j] + bv + *p, 0.f);
    }
  }
}

// ---------------------------------------------------------------------------
extern "C" void kernel_launch(void* const* d_in, const int* in_sizes, int n_in,
                              void* d_out, int out_size, void* d_ws,
                              size_t ws_size, hipStream_t stream) {
  const float*     feat   = (const float*)d_in[0];
  const float*     weight = (const float*)d_in[1];
  const float*     w_comp = (const float*)d_in[2];
  const float*     h_bias = (const float*)d_in[3];
  const float*     loop_w = (const float*)d_in[4];
  const float*     norm   = (const float*)d_in[5];
  const long long* src    = (const long long*)d_in[6];  // int64 per reference
  const long long* dst    = (const long long*)d_in[7];
  float* out  = (float*)d_out;
  float* Wbuf = (float*)d_ws;  // 64*128*128 f32 = 4 MB scratch

  const size_t edgeShmem = (16384 + 16896 + 128) * sizeof(float) + 128 * sizeof(int);
  const size_t loopShmem = (16384 + 16896) * sizeof(float);
  (void)hipFuncSetAttribute((const void*)rgcn_edge_kernel,
                            hipFuncAttributeMaxDynamicSharedMemorySize,
                            (int)edgeShmem);
  (void)hipFuncSetAttribute((const void*)rgcn_loop_kernel,
                            hipFuncAttributeMaxDynamicSharedMemorySize,
                            (int)loopShmem);

  // 1) zero the scatter target
  {
    int n4 = (N_NODES * OUT_FEAT) / 4;  // 3,200,000
    zero_out_kernel<<<(n4 + 255) / 256, 256, 0, stream>>>((float4*)out, n4);
  }
  // 2) basis composition -> Wbuf
  basis_kernel<<<(NUM_RELS * 16384) / 256, 256, 0, stream>>>(w_comp, weight, Wbuf);
  // 3) fused gather + grouped WMMA GEMM + norm + atomic scatter
  rgcn_edge_kernel<<<E_TOTAL / 128, 256, edgeShmem, stream>>>(feat, Wbuf, norm,
                                                             src, dst, out);
  // 4) self-loop WMMA GEMM + bias + ReLU (in place)
  rgcn_loop_kernel<<<(M_TILES + 7) / 8, 256, loopShmem, stream>>>(feat, loop_w,
                                                                 h_bias, out);
}